// lGNN_multiclass_53712861003913
// MI455X (gfx1250) — compile-verified
//
#include <hip/hip_runtime.h>

typedef __attribute__((ext_vector_type(2))) float v2f;
typedef __attribute__((ext_vector_type(4))) float v4f;
typedef __attribute__((ext_vector_type(8))) float v8f;

#define BN_EPS 1e-5f
#define KSPLIT 4

// ---------------------------------------------------------------------------
// Transpose P [N][M] (float2 elems) -> Pt [M][N], LDS-tiled, coalesced.
// ---------------------------------------------------------------------------
struct alignas(8) f2 { float a, b; };

__global__ void k_transpose(const f2* __restrict__ in, f2* __restrict__ out,
                            int N, int M) {
  __shared__ f2 tile[32][33];
  int bx = blockIdx.x * 32;  // m tile
  int by = blockIdx.y * 32;  // n tile
  int tx = threadIdx.x, ty = threadIdx.y;  // 32 x 8
  for (int i = ty; i < 32; i += 8)
    tile[i][tx] = in[(size_t)(by + i) * M + (bx + tx)];
  __syncthreads();
  for (int i = ty; i < 32; i += 8)
    out[(size_t)(bx + i) * N + (by + tx)] = tile[tx][i];
}

// ---------------------------------------------------------------------------
// Prep: B[(m*J+j)][col] = sum_g BN(h[m,g]) * Wsm[(j*Fin+g)*nw + col]
// written directly in WMMA B-fragment packed layout (tiles adjacent per lane
// so the GEMM can fetch both tiles with one global_load_b128):
//   Bp[(((s*32 + lane)*NT + t)*2 + e],  k=4s+2*(lane>>4)+e, col=t*16+(lane&15)
// stats==nullptr -> raw h (first layer, Fin==1).
// ---------------------------------------------------------------------------
__global__ void k_prep(const float* __restrict__ H, const float* __restrict__ stats,
                       const float* __restrict__ gamma, const float* __restrict__ beta,
                       const float* __restrict__ Wsm, float* __restrict__ Bp,
                       int Fin, int J, int NT, int nw, int total) {
  int tid = blockIdx.x * blockDim.x + threadIdx.x;
  if (tid >= total) return;
  int ncols = NT * 16;
  int col = tid % ncols;
  int q   = tid / ncols;
  int m = q / J, j = q % J;
  float val = 0.f;
  if (col < nw) {
    const float* hrow = H + (size_t)m * Fin;
    for (int g = 0; g < Fin; ++g) {
      float h = hrow[g];
      if (stats) {
        float mu = stats[g], var = stats[Fin + g];
        h = gamma[g] * (h - mu) * rsqrtf(var + BN_EPS) + beta[g];
      }
      val += h * Wsm[(j * Fin + g) * nw + col];
    }
  }
  int s = q >> 2, sub = q & 3;
  int lane = ((sub >> 1) << 4) | (col & 15);
  int e = sub & 1;
  int t = col >> 4;
  Bp[((((size_t)s * 32 + lane) * NT + t) * 2) + e] = val;
}

// ---------------------------------------------------------------------------
// GEMM core: one wave = 16 rows x (NT*16) cols, fp32 WMMA 16x16x4.
// Unroll-by-8 body with loop-carried pointers (offsets folded into IOFFSET),
// one unconditional global_prefetch per body, b128 B-fragment loads.
// ---------------------------------------------------------------------------
template <int NT>
__device__ __forceinline__ void gemm_stream(const float* __restrict__ A, int K,
                                            const float* __restrict__ Bp,
                                            int ar, int kofs, int lane,
                                            int split, int S,
                                            v8f& acc0, v8f& acc1) {
  int s0 = ((K >> 2) * split) / S;
  int s1 = ((K >> 2) * (split + 1)) / S;
  const float* Ap = A + (size_t)ar * K + kofs + 4 * s0;
  const float* Bl = Bp + ((size_t)s0 * 32 + lane) * (NT * 2);
  int iters = (s1 - s0) >> 3;  // all step counts divisible by 8
  for (int it = 0; it < iters; ++it) {
    __builtin_prefetch(Ap + 4096, 0, 0);  // 16 KB ahead in this lane's row
#pragma unroll
    for (int u = 0; u < 8; ++u) {
      v2f a = *(const v2f*)(Ap + 4 * u);
      if (NT == 2) {
        v4f bb = *(const v4f*)(Bl + u * 128);
        v2f b0 = { bb[0], bb[1] };
        v2f b1 = { bb[2], bb[3] };
        acc0 = __builtin_amdgcn_wmma_f32_16x16x4_f32(false, a, false, b0,
                                                     (short)0, acc0, false, false);
        acc1 = __builtin_amdgcn_wmma_f32_16x16x4_f32(false, a, false, b1,
                                                     (short)0, acc1, false, false);
      } else {
        v2f b0 = *(const v2f*)(Bl + u * 64);
        acc0 = __builtin_amdgcn_wmma_f32_16x16x4_f32(false, a, false, b0,
                                                     (short)0, acc0, false, false);
      }
    }
    Ap += 32;          // 8 steps * 4 floats
    Bl += NT * 512;    // 8 steps * 32 lanes * NT*2 floats
  }
}

template <int NT>
__global__ void k_gemm(const float* __restrict__ A1, int K1,
                       const float* __restrict__ A2, int K2,
                       const float* __restrict__ B1, const float* __restrict__ B2,
                       float* __restrict__ Zp, int R, int S) {
  int lane = threadIdx.x & 31;
  int wave = threadIdx.x >> 5;
  int row0 = (blockIdx.x * (blockDim.x >> 5) + wave) * 16;
  int split = blockIdx.y;
  int ar = row0 + (lane & 15);
  int kofs = (lane >> 4) << 1;
  v8f acc0 = {};
  v8f acc1 = {};
  gemm_stream<NT>(A1, K1, B1, ar, kofs, lane, split, S, acc0, acc1);
  gemm_stream<NT>(A2, K2, B2, ar, kofs, lane, split, S, acc0, acc1);
  // C layout: VGPR r -> row = row0 + (lane>>4)*8 + r, col = lane&15 (+16 tile1)
  int col = lane & 15;
  int rbase = row0 + ((lane >> 4) << 3);
  const int ncf = NT * 16;
  float* Zw = Zp + (size_t)split * R * ncf + (size_t)rbase * ncf + col;
#pragma unroll
  for (int r = 0; r < 8; ++r) {
    Zw[(size_t)r * ncf] = acc0[r];
    if (NT == 2) Zw[(size_t)r * ncf + 16] = acc1[r];
  }
}

// ---------------------------------------------------------------------------
// Combine K-split partials + bias, ReLU on first nrelu columns.
// ---------------------------------------------------------------------------
__global__ void k_combine(const float* __restrict__ Zp, int R, int ncfull, int S,
                          const float* __restrict__ bias, float* __restrict__ Z,
                          int ncout, int nrelu) {
  int tid = blockIdx.x * blockDim.x + threadIdx.x;
  if (tid >= R * ncout) return;
  int c = tid % ncout, r = tid / ncout;
  float v = bias[c];
  for (int s = 0; s < S; ++s) v += Zp[((size_t)s * R + r) * ncfull + c];
  if (c < nrelu) v = fmaxf(v, 0.f);
  Z[(size_t)r * ncout + c] = v;
}

// ---------------------------------------------------------------------------
// Per-column mean / biased variance over R rows (training-mode BatchNorm).
// ---------------------------------------------------------------------------
__global__ void k_stats(const float* __restrict__ Z, int R, int F,
                        float* __restrict__ stats) {
  int c = blockIdx.x;
  __shared__ float ssum[256];
  __shared__ float ssq[256];
  float s = 0.f, q = 0.f;
  for (int r = threadIdx.x; r < R; r += 256) {
    float v = Z[(size_t)r * F + c];
    s += v;
    q += v * v;
  }
  ssum[threadIdx.x] = s;
  ssq[threadIdx.x] = q;
  __syncthreads();
  for (int o = 128; o > 0; o >>= 1) {
    if ((int)threadIdx.x < o) {
      ssum[threadIdx.x] += ssum[threadIdx.x + o];
      ssq[threadIdx.x] += ssq[threadIdx.x + o];
    }
    __syncthreads();
  }
  if (threadIdx.x == 0) {
    float mu = ssum[0] / R;
    stats[c] = mu;
    stats[F + c] = ssq[0] / R - mu * mu;
  }
}

// ---------------------------------------------------------------------------
extern "C" void kernel_launch(void* const* d_in, const int* in_sizes, int n_in,
                              void* d_out, int out_size, void* d_ws, size_t ws_size,
                              hipStream_t stream) {
  (void)n_in; (void)out_size; (void)ws_size;
  const float* W     = (const float*)d_in[0];
  const float* x     = (const float*)d_in[1];
  const float* Wlg   = (const float*)d_in[2];
  const float* y     = (const float*)d_in[3];
  const float* P     = (const float*)d_in[4];
  const float* wx2x0 = (const float*)d_in[5];
  const float* wy2x0 = (const float*)d_in[6];
  const float* bx0   = (const float*)d_in[7];
  const float* wy2y0 = (const float*)d_in[8];
  const float* wx2y0 = (const float*)d_in[9];
  const float* by0   = (const float*)d_in[10];
  const float* gx0   = (const float*)d_in[11];
  const float* btx0  = (const float*)d_in[12];
  const float* gy0   = (const float*)d_in[13];
  const float* bty0  = (const float*)d_in[14];
  const float* wx2x_m = (const float*)d_in[15];
  const float* wy2x_m = (const float*)d_in[16];
  const float* bx_m   = (const float*)d_in[17];
  const float* wy2y_m = (const float*)d_in[18];
  const float* wx2y_m = (const float*)d_in[19];
  const float* by_m   = (const float*)d_in[20];
  const float* gx_m   = (const float*)d_in[21];
  const float* btx_m  = (const float*)d_in[22];
  const float* gy_m   = (const float*)d_in[23];
  const float* bty_m  = (const float*)d_in[24];
  const float* wlx    = (const float*)d_in[25];
  const float* wly    = (const float*)d_in[26];
  const float* bl     = (const float*)d_in[27];

  const int N = in_sizes[1];            // 2048
  const int M = in_sizes[3];            // 4096
  const int J = 3;
  const int F = in_sizes[5] / J;        // 32
  const int L = in_sizes[17] / F;       // 6
  const int C = in_sizes[27];           // 2

  // ---- workspace carve (floats) ----
  float* ws = (float*)d_ws;
  size_t off = 0;
  auto alloc = [&](size_t n) { float* p = ws + off; off += n; return p; };
  float* Pt     = alloc((size_t)M * N * 2);          // transposed P
  float* zxb[2] = { alloc((size_t)N * F), alloc((size_t)N * F) };
  float* zyb[2] = { alloc((size_t)M * F), alloc((size_t)M * F) };
  float* sxb[2] = { alloc(2 * (size_t)F), alloc(2 * (size_t)F) };
  float* syb[2] = { alloc(2 * (size_t)F), alloc(2 * (size_t)F) };
  float* Bu  = alloc((size_t)N * J * F);
  float* Bv  = alloc((size_t)M * 2 * F);
  float* Bw  = alloc((size_t)M * J * F);
  float* Bz  = alloc((size_t)N * 2 * F);
  float* Zpx = alloc((size_t)KSPLIT * N * F);
  float* Zpy = alloc((size_t)KSPLIT * M * F);
  float* Bul = alloc((size_t)N * J * 16);
  float* Bvl = alloc((size_t)M * 2 * 16);
  float* Zpo = alloc((size_t)KSPLIT * N * 16);

  // ---- one-time transpose of P (layer-invariant) ----
  k_transpose<<<dim3(M / 32, N / 32), dim3(32, 8), 0, stream>>>(
      (const f2*)P, (f2*)Pt, N, M);

  const int PB = 256;
  auto cdiv = [](int a, int b) { return (a + b - 1) / b; };

  for (int b = 0; b <= L; ++b) {
    const float *wxx, *wyx, *bxv, *wyy, *wxy, *byv;
    const float *Hx, *Hy, *stx, *sty;
    const float *gX = nullptr, *bX = nullptr, *gY = nullptr, *bY = nullptr;
    int Fin;
    if (b == 0) {
      wxx = wx2x0; wyx = wy2x0; bxv = bx0; wyy = wy2y0; wxy = wx2y0; byv = by0;
      Hx = x; Hy = y; stx = nullptr; sty = nullptr; Fin = 1;
    } else {
      int l = b - 1;
      wxx = wx2x_m + (size_t)l * J * F * F;
      wyx = wy2x_m + (size_t)l * 2 * F * F;
      bxv = bx_m   + (size_t)l * F;
      wyy = wy2y_m + (size_t)l * J * F * F;
      wxy = wx2y_m + (size_t)l * 2 * F * F;
      byv = by_m   + (size_t)l * F;
      Hx = zxb[(b - 1) & 1]; Hy = zyb[(b - 1) & 1];
      stx = sxb[(b - 1) & 1]; sty = syb[(b - 1) & 1];
      if (b == 1) { gX = gx0; bX = btx0; gY = gy0; bY = bty0; }
      else {
        int p = b - 2;
        gX = gx_m + (size_t)p * F; bX = btx_m + (size_t)p * F;
        gY = gy_m + (size_t)p * F; bY = bty_m + (size_t)p * F;
      }
      Fin = F;
    }
    // B operands for this block (BN folded into prep)
    int tBu = N * J * 2 * 16, tBv = M * 2 * 2 * 16, tBw = M * J * 2 * 16, tBz = N * 2 * 2 * 16;
    k_prep<<<cdiv(tBu, PB), PB, 0, stream>>>(Hx, stx, gX, bX, wxx, Bu, Fin, J, 2, F, tBu);
    k_prep<<<cdiv(tBv, PB), PB, 0, stream>>>(Hy, sty, gY, bY, wyx, Bv, Fin, 2, 2, F, tBv);
    k_prep<<<cdiv(tBw, PB), PB, 0, stream>>>(Hy, sty, gY, bY, wyy, Bw, Fin, J, 2, F, tBw);
    k_prep<<<cdiv(tBz, PB), PB, 0, stream>>>(Hx, stx, gX, bX, wxy, Bz, Fin, 2, 2, F, tBz);
    // big fused GEMMs: zx = W2d@Bu + P2d@Bv ; zy = Wlg2d@Bw + Pt2d@Bz
    k_gemm<2><<<dim3(N / 128, KSPLIT), 256, 0, stream>>>(W, N * J, P, 2 * M, Bu, Bv, Zpx, N, KSPLIT);
    k_gemm<2><<<dim3(M / 128, KSPLIT), 256, 0, stream>>>(Wlg, M * J, Pt, 2 * N, Bw, Bz, Zpy, M, KSPLIT);
    k_combine<<<cdiv(N * F, PB), PB, 0, stream>>>(Zpx, N, 2 * 16, KSPLIT, bxv, zxb[b & 1], F, F / 2);
    k_combine<<<cdiv(M * F, PB), PB, 0, stream>>>(Zpy, M, 2 * 16, KSPLIT, byv, zyb[b & 1], F, F / 2);
    k_stats<<<F, 256, 0, stream>>>(zxb[b & 1], N, F, sxb[b & 1]);
    k_stats<<<F, 256, 0, stream>>>(zyb[b & 1], M, F, syb[b & 1]);
  }

  // ---- final linear layer (x branch): out = W2d@Bul + P2d@Bvl + bl ----
  int li = L & 1;
  const float* gXf = gx_m + (size_t)(L - 1) * F;
  const float* bXf = btx_m + (size_t)(L - 1) * F;
  const float* gYf = gy_m + (size_t)(L - 1) * F;
  const float* bYf = bty_m + (size_t)(L - 1) * F;
  int tBul = N * J * 16, tBvl = M * 2 * 16;
  k_prep<<<cdiv(tBul, PB), PB, 0, stream>>>(zxb[li], sxb[li], gXf, bXf, wlx, Bul, F, J, 1, C, tBul);
  k_prep<<<cdiv(tBvl, PB), PB, 0, stream>>>(zyb[li], syb[li], gYf, bYf, wly, Bvl, F, 2, 1, C, tBvl);
  k_gemm<1><<<dim3(N / 128, KSPLIT), 256, 0, stream>>>(W, N * J, P, 2 * M, Bul, Bvl, Zpo, N, KSPLIT);
  k_combine<<<cdiv(N * C, PB), PB, 0, stream>>>(Zpo, N, 16, KSPLIT, bl, (float*)d_out, C, 0);
}